// GlobalLocalContrastiveLoss_38439957299270
// MI455X (gfx1250) — compile-verified
//
#include <hip/hip_runtime.h>
#include <math.h>

// ---------------------------------------------------------------------------
// GlobalLocalContrastiveLoss on MI455X (gfx1250)
//
// Virtual negative matrix N = [z_t flattened (2048 rows) ; memory_queue
// (65536 rows)], each row D=512 f32.  Compute S = (g/T) @ N^T with
// V_WMMA_F32_16X16X4_F32 (two 16x16 f32 accumulators cover all 32 anchors),
// mask the m==clip diagonal on z_t tiles (those entries are `pos`), and fold
// everything into a streaming online logsumexp.  Second tiny kernel combines
// per-workgroup partials and reduces mean(softplus(neg_lse - pos)).
//
// Occupancy: 264 WGs x 8 waves = 2112 wave streams, exactly 2 column tiles
// each (4224 tiles).  ~98 KB LDS per WG allows co-resident WGs per WGP.
// ---------------------------------------------------------------------------

typedef float v2f __attribute__((ext_vector_type(2)));
typedef float v8f __attribute__((ext_vector_type(8)));

#define INV_T      14.285714f   // 1/0.07
#define D_DIM      512
#define B_ANCH     32
#define L_FRAMES   64
#define NZ_ROWS    2048         // B * L
#define LDS_ROW    516          // dword row stride for g in LDS (16B aligned, bank-spread)
#define NWG        264
#define BLOCK_SZ   256
#define WAVES_WG   8

__global__ __launch_bounds__(BLOCK_SZ)
void gl_contrastive_gemm_lse(const float* __restrict__ g,
                             const float* __restrict__ z,      // [2048][512]
                             const float* __restrict__ q,      // [K][512]
                             int K,
                             float* __restrict__ pos,          // [2048]
                             float* __restrict__ pmax,         // [NWG][32]
                             float* __restrict__ psum) {       // [NWG][32]
    __shared__ float gl[B_ANCH * LDS_ROW];     // g * (1/T), padded rows
    __shared__ float redM[B_ANCH * 128];       // per-(wave,lane) partial max keyed by m
    __shared__ float redS[B_ANCH * 128];       // per-(wave,lane) partial sumexp

    const int tid = threadIdx.x;

    // --- stage g/T into LDS (float4 loads, conflict-free padded rows) ------
    for (int i = tid; i < B_ANCH * (D_DIM / 4); i += BLOCK_SZ) {
        const int row = i >> 7;            // 128 float4 per row
        const int c4  = i & 127;
        const float4 v = ((const float4*)g)[i];
        float* dst = &gl[row * LDS_ROW + c4 * 4];
        dst[0] = v.x * INV_T;
        dst[1] = v.y * INV_T;
        dst[2] = v.z * INV_T;
        dst[3] = v.w * INV_T;
    }
    __syncthreads();

    const int wave = tid >> 5;
    const int lane = tid & 31;
    const int h    = lane >> 4;     // half-wave: selects K sub-pair in A/B frags
    const int ln   = lane & 15;

    // A-fragment LDS base pointers (M=0..15 and M=16..31 blocks)
    const float* a0base = &gl[ln * LDS_ROW + 2 * h];
    const float* a1base = &gl[(16 + ln) * LDS_ROW + 2 * h];

    // online logsumexp state: slot r -> m = r + 8h (acc0), slot 8+r -> m = 16 + r + 8h (acc1)
    float rmax[16], rsum[16];
#pragma unroll
    for (int j = 0; j < 16; ++j) { rmax[j] = -INFINITY; rsum[j] = 0.0f; }

    const int ntiles   = (NZ_ROWS + K) >> 4;
    const int nstreams = gridDim.x * WAVES_WG;
    const int stream   = blockIdx.x * WAVES_WG + wave;

    for (int t = stream; t < ntiles; t += nstreams) {
        const int  c0  = t << 4;
        const bool isZ = (c0 < NZ_ROWS);
        // per-lane B-fragment pointer: row = negative index n, col = K position
        const float* bptr = (isZ ? (z + (size_t)(c0 + ln) * D_DIM)
                                 : (q + (size_t)(c0 - NZ_ROWS + ln) * D_DIM)) + 2 * h;

        v8f acc0 = {};
        v8f acc1 = {};
#pragma unroll 8
        for (int kk = 0; kk < D_DIM; kk += 4) {
            v2f bf = *(const v2f*)(bptr + kk);        // global_load_b64 (B 4x16 frag)
            v2f a0 = *(const v2f*)(a0base + kk);      // ds_load_b64    (A 16x4, M=0..15)
            v2f a1 = *(const v2f*)(a1base + kk);      // ds_load_b64    (A 16x4, M=16..31)
            acc0 = __builtin_amdgcn_wmma_f32_16x16x4_f32(false, a0, false, bf,
                                                         (short)0, acc0, false, false);
            acc1 = __builtin_amdgcn_wmma_f32_16x16x4_f32(false, a1, false, bf,
                                                         (short)0, acc1, false, false);
        }

        // epilogue: C layout -> lane covers n = c0+ln; vgpr r -> m = r + 8h (+16 for acc1)
        const int nglob = c0 + ln;
        const int cclip = nglob >> 6;   // clip index of this z_t column
        const int lidx  = nglob & 63;
#pragma unroll
        for (int r = 0; r < 8; ++r) {
            {
                const int   m = r + 8 * h;
                const float s = acc0[r];
                if (isZ && m == cclip) {
                    pos[m * L_FRAMES + lidx] = s;     // diagonal -> positive logit
                } else {
                    const float nm = fmaxf(rmax[r], s);
                    rsum[r] = rsum[r] * __expf(rmax[r] - nm) + __expf(s - nm);
                    rmax[r] = nm;
                }
            }
            {
                const int   m = 16 + r + 8 * h;
                const float s = acc1[r];
                if (isZ && m == cclip) {
                    pos[m * L_FRAMES + lidx] = s;
                } else {
                    const int j = 8 + r;
                    const float nm = fmaxf(rmax[j], s);
                    rsum[j] = rsum[j] * __expf(rmax[j] - nm) + __expf(s - nm);
                    rmax[j] = nm;
                }
            }
        }
    }

    // --- scatter per-lane partials into LDS keyed by anchor m --------------
    const int idx = wave * 16 + ln;
#pragma unroll
    for (int r = 0; r < 8; ++r) {
        const int m0 = r + 8 * h;
        redM[m0 * 128 + idx] = rmax[r];
        redS[m0 * 128 + idx] = rsum[r];
        const int m1 = 16 + r + 8 * h;
        redM[m1 * 128 + idx] = rmax[8 + r];
        redS[m1 * 128 + idx] = rsum[8 + r];
    }
    __syncthreads();

    if (tid < B_ANCH) {
        float M = -INFINITY, S = 0.0f;
        for (int i = 0; i < 128; ++i) {
            const float mi = redM[tid * 128 + i];
            const float si = redS[tid * 128 + i];
            const float nm = fmaxf(M, mi);
            S = S * __expf(M - nm) + si * __expf(mi - nm);
            M = nm;
        }
        pmax[blockIdx.x * B_ANCH + tid] = M;
        psum[blockIdx.x * B_ANCH + tid] = S;
    }
}

__global__ __launch_bounds__(BLOCK_SZ)
void gl_contrastive_finalize(const float* __restrict__ pos,
                             const float* __restrict__ pmax,
                             const float* __restrict__ psum,
                             int nwg,
                             float* __restrict__ out) {
    __shared__ float lse[B_ANCH];
    __shared__ float red[BLOCK_SZ];
    const int tid = threadIdx.x;

    if (tid < B_ANCH) {
        float M = -INFINITY, S = 0.0f;
        for (int i = 0; i < nwg; ++i) {
            const float mi = pmax[i * B_ANCH + tid];
            const float si = psum[i * B_ANCH + tid];
            const float nm = fmaxf(M, mi);
            S = S * __expf(M - nm) + si * __expf(mi - nm);
            M = nm;
        }
        lse[tid] = M + __logf(S);   // neg_lse[b]
    }
    __syncthreads();

    float acc = 0.0f;
    for (int i = tid; i < NZ_ROWS; i += BLOCK_SZ) {
        const float p = pos[i];
        const float x = lse[i >> 6] - p;                       // neg_lse - pos
        acc += fmaxf(x, 0.0f) + log1pf(__expf(-fabsf(x)));     // softplus(x)
    }
    red[tid] = acc;
    __syncthreads();
    for (int s = BLOCK_SZ / 2; s > 0; s >>= 1) {
        if (tid < s) red[tid] += red[tid + s];
        __syncthreads();
    }
    if (tid == 0) out[0] = red[0] / (float)NZ_ROWS;
}

extern "C" void kernel_launch(void* const* d_in, const int* in_sizes, int n_in,
                              void* d_out, int out_size, void* d_ws, size_t ws_size,
                              hipStream_t stream) {
    (void)n_in; (void)out_size; (void)ws_size;
    const float* g  = (const float*)d_in[0];   // [32, 512]
    const float* zt = (const float*)d_in[1];   // [32, 64, 512]
    const float* mq = (const float*)d_in[2];   // [K, 512]
    const int K = in_sizes[2] / D_DIM;         // 65536 for the reference shapes

    float* ws   = (float*)d_ws;
    float* pos  = ws;                          // 2048 floats
    float* pmax = ws + NZ_ROWS;                // NWG*32 floats
    float* psum = pmax + NWG * B_ANCH;         // NWG*32 floats

    gl_contrastive_gemm_lse<<<NWG, BLOCK_SZ, 0, stream>>>(g, zt, mq, K, pos, pmax, psum);
    gl_contrastive_finalize<<<1, BLOCK_SZ, 0, stream>>>(pos, pmax, psum, NWG,
                                                        (float*)d_out);
}